// LTC_51153060496052
// MI455X (gfx1250) — compile-verified
//
#include <hip/hip_runtime.h>

#define B_SZ   256
#define L_SZ   1024
#define IN_SZ  64
#define UNITS  64
#define UNF    6
#define LOG2E  1.4426950408889634f
#define LN2    0.6931471805599453f

// softplus(x) = ln2 * log2(1 + exp2(x * log2e))
__device__ __forceinline__ float softplus_f(float x) {
    float e = __builtin_amdgcn_exp2f(x * LOG2E);
    return LN2 * __builtin_amdgcn_logf(1.0f + e);
}

// Sum across the 4 lanes of a quad using DPP quad_perm (pure VALU, no DS pipe).
// xor1 = quad_perm [1,0,3,2] = 0xB1 ; xor2 = quad_perm [2,3,0,1] = 0x4E
__device__ __forceinline__ float quad_reduce_add(float v) {
    int s  = __float_as_int(v);
    int p1 = __builtin_amdgcn_update_dpp(0, s, 0xB1, 0xF, 0xF, false);
    v += __int_as_float(p1);
    s  = __float_as_int(v);
    int p2 = __builtin_amdgcn_update_dpp(0, s, 0x4E, 0xF, 0xF, false);
    v += __int_as_float(p2);
    return v;
}

// wave-0 helper: async-copy 64 floats (256B) of x into LDS, one b64 per lane.
// CDNA5 GLOBAL_LOAD_ASYNC_TO_LDS (tracked by ASYNCcnt), GVS addressing:
//   global address is SADDR plus the VADDR offset, LDS address comes from VDST
__device__ __forceinline__ void async_copy_x(const float* gsrc, unsigned lds_base) {
    const unsigned lane = threadIdx.x & 31u;
    const unsigned ldsa = lds_base + lane * 8u;
    const unsigned voff = lane * 8u;
    asm volatile("global_load_async_to_lds_b64 %0, %1, %2"
                 :: "v"(ldsa), "v"(voff), "s"(gsrc)
                 : "memory");
}

__global__ __launch_bounds__(256) void ltc_scan_kernel(
    const float* __restrict__ x,
    const float* __restrict__ gleak, const float* __restrict__ vleak,
    const float* __restrict__ cm,
    const float* __restrict__ sigma, const float* __restrict__ mu,
    const float* __restrict__ w,     const float* __restrict__ erev,
    const float* __restrict__ ss,    const float* __restrict__ sm,
    const float* __restrict__ sw,    const float* __restrict__ serev,
    const float* __restrict__ iw,    const float* __restrict__ ib,
    const float* __restrict__ ow,    const float* __restrict__ ob,
    float* __restrict__ readout, float* __restrict__ hT)
{
    __shared__ float2 sAB[UNITS * IN_SZ];          // sensory exp2 coeffs, [j][i], 32KB
    __shared__ __align__(16) float vbuf[2][UNITS]; // ping-pong hidden state
    __shared__ __align__(16) float xbuf[2][IN_SZ]; // double-buffered x row

    const int tid = threadIdx.x;
    const int b   = blockIdx.x;
    const int j   = tid >> 2;   // output unit (column), 0..63
    const int g   = tid & 3;    // quad id -> row group, i in [16g, 16g+16)
    const int i0  = g * 16;

    // ---------------- one-time setup: fold params ----------------
    float rA[16], rBc[16], rWn[16], rWd[16], sWn[16], sWd[16];
#pragma unroll
    for (int k = 0; k < 16; ++k) {
        const int i   = i0 + k;
        const int idx = i * UNITS + j;
        // exponent of sigmoid: exp2(A*v + B), A=-log2e*sigma, B=log2e*sigma*mu
        float sg = sigma[idx];
        float m  = mu[idx];
        float wv = softplus_f(w[idx]);
        rA[k]  = -LOG2E * sg;
        rBc[k] =  LOG2E * sg * m;
        rWd[k] =  wv;
        rWn[k] =  wv * erev[idx];
        // sensory: input affine iw*x+ib folded into coefficients
        float g2  = ss[idx];
        float m2  = sm[idx];
        float wv2 = softplus_f(sw[idx]);
        sWd[k] = wv2;
        sWn[k] = wv2 * serev[idx];
        float2 ab;
        ab.x = -LOG2E * g2 * iw[i];
        ab.y =  LOG2E * g2 * (m2 - ib[i]);
        sAB[j * IN_SZ + i] = ab;
    }
    // per-unit scalars
    const float gl   = softplus_f(gleak[j]);
    const float glvl = gl * vleak[j];
    const float cmt  = softplus_f(cm[j]) * (float)UNF;
    const float denc = cmt + gl + 1e-8f;     // EPS folded in
    const float owj  = ow[j], obj = ob[j];

    if (g == 0) vbuf[0][j] = 0.0f;           // h0 = 0

    const float*   xrow  = x + (size_t)b * (size_t)L_SZ * IN_SZ;
    const unsigned xbase = (unsigned)(size_t)(&xbuf[0][0]);

    if (tid < 32) async_copy_x(xrow, xbase);          // prefetch t=0

    // ---------------- sequential scan over L ----------------
#pragma unroll 1
    for (int t = 0; t < L_SZ; ++t) {
        if (tid < 32) {
            asm volatile("s_wait_asynccnt 0x0" ::: "memory");  // x[t] landed
            if (t + 1 < L_SZ)                                   // prefetch t+1
                async_copy_x(xrow + (size_t)(t + 1) * IN_SZ,
                             xbase + (unsigned)(((t + 1) & 1) * (IN_SZ * 4)));
        }
        __syncthreads();   // x[t] + previous v visible to all waves

        // ----- sensory phase (constant across unfolds) -----
        const int xb = t & 1;
        float xv[16];
        {
            const float4* xp = (const float4*)(&xbuf[xb][0]);
            float4 x0 = xp[g * 4 + 0], x1 = xp[g * 4 + 1];
            float4 x2 = xp[g * 4 + 2], x3 = xp[g * 4 + 3];
            xv[0]=x0.x; xv[1]=x0.y; xv[2]=x0.z; xv[3]=x0.w;
            xv[4]=x1.x; xv[5]=x1.y; xv[6]=x1.z; xv[7]=x1.w;
            xv[8]=x2.x; xv[9]=x2.y; xv[10]=x2.z; xv[11]=x2.w;
            xv[12]=x3.x; xv[13]=x3.y; xv[14]=x3.z; xv[15]=x3.w;
        }
        float sn = 0.0f, sd = 0.0f;
#pragma unroll
        for (int k = 0; k < 16; ++k) {
            float2 ab = sAB[j * IN_SZ + i0 + k];
            float e = __builtin_amdgcn_exp2f(__builtin_fmaf(ab.x, xv[k], ab.y));
            float r = __builtin_amdgcn_rcpf(1.0f + e);
            sn = __builtin_fmaf(sWn[k], r, sn);
            sd = __builtin_fmaf(sWd[k], r, sd);
        }
        sn = quad_reduce_add(sn);
        sd = quad_reduce_add(sd);

        // ----- 6 ODE unfolds, LDS ping-pong state -----
#pragma unroll
        for (int u = 0; u < UNF; ++u) {
            const int rp = u & 1;            // read buffer (starts at 0)
            float vv[16];
            {
                const float4* vp = (const float4*)(&vbuf[rp][0]);
                float4 a0 = vp[g * 4 + 0], a1 = vp[g * 4 + 1];
                float4 a2 = vp[g * 4 + 2], a3 = vp[g * 4 + 3];
                vv[0]=a0.x; vv[1]=a0.y; vv[2]=a0.z; vv[3]=a0.w;
                vv[4]=a1.x; vv[5]=a1.y; vv[6]=a1.z; vv[7]=a1.w;
                vv[8]=a2.x; vv[9]=a2.y; vv[10]=a2.z; vv[11]=a2.w;
                vv[12]=a3.x; vv[13]=a3.y; vv[14]=a3.z; vv[15]=a3.w;
            }
            float rn = 0.0f, rd = 0.0f;
#pragma unroll
            for (int k = 0; k < 16; ++k) {
                float e = __builtin_amdgcn_exp2f(__builtin_fmaf(rA[k], vv[k], rBc[k]));
                float r = __builtin_amdgcn_rcpf(1.0f + e);
                rn = __builtin_fmaf(rWn[k], r, rn);
                rd = __builtin_fmaf(rWd[k], r, rd);
            }
            rn = quad_reduce_add(rn);
            rd = quad_reduce_add(rd);

            if (g == 0) {
                float vj  = vbuf[rp][j];
                float num = __builtin_fmaf(cmt, vj, glvl) + rn + sn;
                float den = denc + rd + sd;
                vbuf[rp ^ 1][j] = num * __builtin_amdgcn_rcpf(den);
            }
            __syncthreads();                 // publish v for next unfold
        }

        // after an even number of flips the state is back in vbuf[0]
        if (g == 0) {
            float vj = vbuf[0][j];
            readout[((size_t)b * L_SZ + t) * UNITS + j] = __builtin_fmaf(vj, owj, obj);
            if (t == L_SZ - 1) hT[b * UNITS + j] = vj;
        }
    }
}

extern "C" void kernel_launch(void* const* d_in, const int* in_sizes, int n_in,
                              void* d_out, int out_size, void* d_ws, size_t ws_size,
                              hipStream_t stream) {
    const float* x     = (const float*)d_in[0];
    const float* gleak = (const float*)d_in[1];
    const float* vleak = (const float*)d_in[2];
    const float* cm    = (const float*)d_in[3];
    const float* sigma = (const float*)d_in[4];
    const float* mu    = (const float*)d_in[5];
    const float* w     = (const float*)d_in[6];
    const float* erev  = (const float*)d_in[7];
    const float* ss    = (const float*)d_in[8];
    const float* sm    = (const float*)d_in[9];
    const float* sw    = (const float*)d_in[10];
    const float* serev = (const float*)d_in[11];
    const float* iw    = (const float*)d_in[12];
    const float* ib    = (const float*)d_in[13];
    const float* ow    = (const float*)d_in[14];
    const float* ob    = (const float*)d_in[15];

    float* out = (float*)d_out;
    float* hT  = out + (size_t)B_SZ * L_SZ * UNITS;   // readout first, then hT

    ltc_scan_kernel<<<dim3(B_SZ), dim3(256), 0, stream>>>(
        x, gleak, vleak, cm, sigma, mu, w, erev,
        ss, sm, sw, serev, iw, ib, ow, ob, out, hT);
}